// rec_13099650253518
// MI455X (gfx1250) — compile-verified
//
#include <hip/hip_runtime.h>

#define SZ      1024
#define HALF    512
#define PATCH   512
#define NBATCH  16
#define NPIX    (SZ * SZ)        // 1,048,576 canvas pixels
#define PPIX    (PATCH * PATCH)  // 262,144 patch pixels
#define NB_RED  1024             // blocks in per-batch diff reduction
#define NTHR    256              // NB_RED*NTHR == NPIX/4 exactly (int4 per thread)

typedef __attribute__((ext_vector_type(2))) float v2f;
typedef __attribute__((ext_vector_type(8))) float v8f;

// -------- wave32 sum via the matrix pipe -----------------------------------
// A (16x4 f32, 2 VGPRs/lane): lane m in [0,16) holds A[m][0]=s, A[m][1]=0;
// lane m+16 holds A[m][2]=s, A[m][3]=0.  B = all-ones (4x16).
// D[m][n] = sum_k A[m][k] = s_m + s_{m+16}  (replicated over n).
// Lane l's accumulators d[0..7] cover M = 8*(l/16)..8*(l/16)+7 at N=l%16, so
// t = sum(d[0..7]);  t += shfl_xor(t,16)  ==> every lane holds the full
// 32-lane sum, computed entirely in f32 (no precision loss).
__device__ __forceinline__ float wave_sum_wmma(float s) {
    v2f a;    a[0] = s;    a[1] = 0.0f;
    v2f ones; ones[0] = 1.0f; ones[1] = 1.0f;
    v8f c = {};
    c = __builtin_amdgcn_wmma_f32_16x16x4_f32(
            /*neg_a=*/false, a, /*neg_b=*/false, ones,
            /*c_mod=*/(short)0, c, /*reuse_a=*/false, /*reuse_b=*/false);
    float t = ((c[0] + c[1]) + (c[2] + c[3])) + ((c[4] + c[5]) + (c[6] + c[7]));
    t += __shfl_xor(t, 16, 32);
    return t;
}

// -------- kernel 1: polar->cartesian index map (batch independent) ---------
__global__ void build_map_kernel(int* __restrict__ map) {
    int p = blockIdx.x * blockDim.x + threadIdx.x;
    if (p >= NPIX) return;
    int row = p >> 10;
    int col = p & (SZ - 1);
    const float dAng = 3.0679615757712823e-03f;   // float32(pi/1024)
    float r   = (float)(HALF - col);
    float ang = (float)row * dAng;
    float x = r * cosf(ang);
    float y = r * sinf(ang);
    int i = (int)rintf(x + 512.0f) - 1;           // rint == half-to-even, same as jnp.round
    int j = (int)rintf(y + 512.0f) - 1;
    i = min(max(i, 0), SZ - 1);
    j = min(max(j, 0), SZ - 1);
    map[p] = j * SZ + i;
}

// -------- kernel 2: zero both rec grids (b128 stores) ----------------------
__global__ void zero_rec_kernel(int4* __restrict__ p) {
    int i = blockIdx.x * blockDim.x + threadIdx.x;   // 2*NPIX/4 elements
    p[i] = make_int4(0, 0, 0, 0);
}

// -------- kernel 3: scatter-max one batch's sr & hr patches ----------------
// Signed-int atomic max on float bits is exact here because dst init is 0:
// non-negative floats order as their int bits; negative floats have negative
// int bits and correctly lose to 0 (reference result is max(0, values)).
__global__ void scatter_max_kernel(const float* __restrict__ sr,
                                   const float* __restrict__ hr,
                                   const int*   __restrict__ row0,
                                   const int*   __restrict__ col0,
                                   const int*   __restrict__ map,
                                   int* __restrict__ srRec,
                                   int* __restrict__ hrRec,
                                   int b) {
    int idx = blockIdx.x * blockDim.x + threadIdx.x;
    if (idx >= PPIX) return;
    int h = idx >> 9;
    int w = idx & (PATCH - 1);
    int row = row0[b] + h;              // row0,col0 in [0,512) -> always in-bounds
    int col = col0[b] + w;
    int lin = map[row * SZ + col];
    float vs = sr[(size_t)b * PPIX + idx];
    float vh = hr[(size_t)b * PPIX + idx];
    atomicMax(&srRec[lin], __float_as_int(vs));
    atomicMax(&hrRec[lin], __float_as_int(vh));
}

// -------- kernel 4: |sr_rec - hr_rec| partial sums for one batch -----------
// NB_RED*NTHR threads, exactly one int4 (b128) load pair per thread:
// no loop, no bounds check, EXEC all-ones at the WMMA.
__global__ void diff_reduce_kernel(const int4* __restrict__ srRec,
                                   const int4* __restrict__ hrRec,
                                   float* __restrict__ partial,
                                   int slot) {
    int p = blockIdx.x * blockDim.x + threadIdx.x;   // [0, NPIX/4)
    int4 a4 = srRec[p];                              // stored bits are 0 or float >= 0
    int4 b4 = hrRec[p];
    float s = (fabsf(__int_as_float(a4.x) - __int_as_float(b4.x)) +
               fabsf(__int_as_float(a4.y) - __int_as_float(b4.y))) +
              (fabsf(__int_as_float(a4.z) - __int_as_float(b4.z)) +
               fabsf(__int_as_float(a4.w) - __int_as_float(b4.w)));

    float t = wave_sum_wmma(s);

    __shared__ float wsum[NTHR / 32];
    int lane = threadIdx.x & 31;
    int wid  = threadIdx.x >> 5;
    if (lane == 0) wsum[wid] = t;
    __syncthreads();
    if (threadIdx.x == 0) {
        float bs = 0.0f;
        #pragma unroll
        for (int i = 0; i < NTHR / 32; ++i) bs += wsum[i];
        partial[slot * NB_RED + blockIdx.x] = bs;
    }
}

// -------- kernel 5: final deterministic sum of all partials ----------------
__global__ void final_reduce_kernel(const float* __restrict__ partial,
                                    float* __restrict__ out) {
    // NBATCH*NB_RED = 16384 partials; 256 threads, exactly 64 iters each.
    float s = 0.0f;
    for (int p = threadIdx.x; p < NBATCH * NB_RED; p += NTHR)
        s += partial[p];
    float t = wave_sum_wmma(s);

    __shared__ float wsum[NTHR / 32];
    int lane = threadIdx.x & 31;
    int wid  = threadIdx.x >> 5;
    if (lane == 0) wsum[wid] = t;
    __syncthreads();
    if (threadIdx.x == 0) {
        float total = 0.0f;
        #pragma unroll
        for (int i = 0; i < NTHR / 32; ++i) total += wsum[i];
        out[0] = total * (1.0f / (float)((size_t)NBATCH * NPIX));  // 2^-24, exact
    }
}

extern "C" void kernel_launch(void* const* d_in, const int* in_sizes, int n_in,
                              void* d_out, int out_size, void* d_ws, size_t ws_size,
                              hipStream_t stream) {
    (void)in_sizes; (void)n_in; (void)out_size; (void)ws_size;

    const float* sr   = (const float*)d_in[0];
    const float* hr   = (const float*)d_in[1];
    const int*   row0 = (const int*)d_in[2];
    const int*   col0 = (const int*)d_in[3];
    float* out = (float*)d_out;

    // Workspace layout (~12.3 MB): map | srRec | hrRec | partials
    char* ws = (char*)d_ws;
    int*   map     = (int*)ws;                                   // NPIX ints
    int*   srRec   = (int*)(ws + (size_t)NPIX * sizeof(int));    // NPIX ints
    int*   hrRec   = srRec + NPIX;                               // NPIX ints (contiguous)
    float* partial = (float*)(hrRec + NPIX);                     // NBATCH*NB_RED floats

    build_map_kernel<<<NPIX / NTHR, NTHR, 0, stream>>>(map);

    for (int b = 0; b < NBATCH; ++b) {
        zero_rec_kernel<<<(2 * NPIX / 4) / NTHR, NTHR, 0, stream>>>((int4*)srRec);
        scatter_max_kernel<<<PPIX / NTHR, NTHR, 0, stream>>>(
            sr, hr, row0, col0, map, srRec, hrRec, b);
        diff_reduce_kernel<<<NB_RED, NTHR, 0, stream>>>(
            (const int4*)srRec, (const int4*)hrRec, partial, b);
    }

    final_reduce_kernel<<<1, NTHR, 0, stream>>>(partial, out);
}